// MultiBranchAutoregressiveDecoder_71176198029553
// MI455X (gfx1250) — compile-verified
//
#include <hip/hip_runtime.h>

#define TC   288
#define TF   12
#define DIM  16
#define HEADS 4
#define HD   4
#define NL_  2
#define LPAD 304
#define EPS_ 1e-5f
#define NTHREADS 256

typedef float v2f __attribute__((ext_vector_type(2)));
typedef float v8f __attribute__((ext_vector_type(8)));

// ---- LDS layout (floats) ----
#define OFF_X    0                         // 304*16 activations
#define OFF_Q    (OFF_X + LPAD*DIM)        // q / attn-out / ffn-out tmp
#define OFF_K    (OFF_Q + LPAD*DIM)        // K cache, 2 layers
#define OFF_V    (OFF_K + NL_*LPAD*DIM)    // V cache, 2 layers
#define OFF_WSC  (OFF_V + NL_*LPAD*DIM)    // 8 waves * 16x64 scratch (also outproj tmp)
#define OFF_SM   (OFF_WSC + 8*1024)        // small vectors
#define SM_VAL  0
#define SM_XIN  16
#define SM_Q1   32
#define SM_ATT  48
#define SM_TMP  64
#define SM_H64  80
#define TOTAL_F (OFF_SM + 160)             // 37536 floats = 150144 bytes

// A fragment (16x4, row-major X[m][k]) or B fragment (4x16 as W[n][k], row-major W):
// lane l supplies the float2 at (row0 + (l&15), k0 + 2*(l>>4)).
__device__ __forceinline__ v2f ld_frag(const float* base, int row0, int k0, int ld) {
  int l = threadIdx.x & 31;
  const float* p = base + (row0 + (l & 15)) * ld + (k0 + 2 * (l >> 4));
  v2f r; r.x = p[0]; r.y = p[1];
  return r;
}

__device__ __forceinline__ v8f wmma4(v2f a, v2f b, v8f c) {
  // D = A(16x4, f32) * B(4x16, f32) + C(16x16, f32)
  return __builtin_amdgcn_wmma_f32_16x16x4_f32(false, a, false, b, (short)0, c, false, false);
}

__device__ __forceinline__ float gsum16(float v) {
  v += __shfl_xor(v, 8, 32);
  v += __shfl_xor(v, 4, 32);
  v += __shfl_xor(v, 2, 32);
  v += __shfl_xor(v, 1, 32);
  return v;
}

__device__ __forceinline__ void ln16(float* v, const float* s, const float* b) {
  float mean = 0.f;
#pragma unroll
  for (int c = 0; c < 16; ++c) mean += v[c];
  mean *= 0.0625f;
  float var = 0.f;
#pragma unroll
  for (int c = 0; c < 16; ++c) { float d = v[c] - mean; var += d * d; }
  var *= 0.0625f;
  float rs = rsqrtf(var + EPS_);
#pragma unroll
  for (int c = 0; c < 16; ++c) v[c] = (v[c] - mean) * rs * s[c] + b[c];
}

__global__ void __launch_bounds__(NTHREADS)
mbar_decoder_kernel(const float* __restrict__ g_ctx, const float* __restrict__ g_tgt,
                    const float* __restrict__ g_pos,
                    const float* __restrict__ g_inw, const float* __restrict__ g_inb,
                    const float* __restrict__ g_outw, const float* __restrict__ g_outb,
                    const float* __restrict__ g_l1s, const float* __restrict__ g_l1b,
                    const float* __restrict__ g_l2s, const float* __restrict__ g_l2b,
                    const float* __restrict__ g_l3s, const float* __restrict__ g_l3b,
                    const float* __restrict__ g_w1, const float* __restrict__ g_b1,
                    const float* __restrict__ g_w2, const float* __restrict__ g_b2,
                    const float* __restrict__ g_pw, const float* __restrict__ g_pb,
                    float* __restrict__ g_out) {
  extern __shared__ __align__(16) float smem[];
  float* s_x   = smem + OFF_X;
  float* s_q   = smem + OFF_Q;
  float* s_kc  = smem + OFF_K;
  float* s_vc  = smem + OFF_V;
  float* s_wsc = smem + OFF_WSC;
  float* s_val = smem + OFF_SM + SM_VAL;
  float* s_xin = smem + OFF_SM + SM_XIN;
  float* s_q1  = smem + OFF_SM + SM_Q1;
  float* s_att = smem + OFF_SM + SM_ATT;
  float* s_tmp = smem + OFF_SM + SM_TMP;
  float* s_h64 = smem + OFF_SM + SM_H64;

  const int tid  = threadIdx.x;
  const int lane = tid & 31;
  const int wave = tid >> 5;
  const int b    = blockIdx.x;
  const float* ctx = g_ctx + (size_t)b * TC * DIM;
  const float* tgt = g_tgt + (size_t)b * TF * DIM;

  // x = ctx + pos_emb
  for (int i = tid; i < TC * DIM; i += NTHREADS) s_x[i] = ctx[i] + g_pos[i];
  __syncthreads();

  // =================== Prefill over 288 context tokens ===================
  for (int l = 0; l < NL_; ++l) {
    const float* Wi = g_inw  + l * 48 * DIM;
    const float* Bi = g_inb  + l * 48;
    const float* Wo = g_outw + l * DIM * DIM;
    const float* Bo = g_outb + l * DIM;
    const float* W1 = g_w1   + l * 64 * DIM;
    const float* B1 = g_b1   + l * 64;
    const float* W2 = g_w2   + l * DIM * 64;
    const float* B2 = g_b2   + l * DIM;
    float* Kc = s_kc + l * LPAD * DIM;
    float* Vc = s_vc + l * LPAD * DIM;

    // ---- QKV projection: (288x16) @ (16x48)^T via WMMA f32 16x16x4 ----
    for (int t5 = wave; t5 < 18 * 3; t5 += 8) {
      int mt = t5 / 3, nt = t5 % 3;
      int m0 = mt * 16;
      v8f acc = {0, 0, 0, 0, 0, 0, 0, 0};
#pragma unroll
      for (int k0 = 0; k0 < 16; k0 += 4)
        acc = wmma4(ld_frag(s_x, m0, k0, DIM), ld_frag(Wi, nt * 16, k0, DIM), acc);
      int col = lane & 15;
      float bias = Bi[nt * 16 + col];
      float* dst = (nt == 0) ? s_q : ((nt == 1) ? Kc : Vc);
      int rb = m0 + 8 * (lane >> 4);
#pragma unroll
      for (int g = 0; g < 8; ++g) dst[(rb + g) * DIM + col] = acc[g] + bias;
    }
    __syncthreads();

    // ---- Causal attention (head_dim=4): VALU online softmax ----
    for (int it = tid; it < TC * HEADS; it += NTHREADS) {
      int p = it >> 2, h = it & 3;
      const float4 qv = *(const float4*)(s_q + p * DIM + h * HD);
      float m = -1e30f, ls = 0.f, o0 = 0.f, o1 = 0.f, o2 = 0.f, o3 = 0.f;
      const float* Kh = Kc + h * HD;
      const float* Vh = Vc + h * HD;
      for (int j = 0; j <= p; ++j) {
        float4 kv = *(const float4*)(Kh + j * DIM);
        float s = 0.5f * (qv.x * kv.x + qv.y * kv.y + qv.z * kv.z + qv.w * kv.w);
        float mn = fmaxf(m, s);
        float cc = __expf(m - mn), e = __expf(s - mn);
        float4 vv = *(const float4*)(Vh + j * DIM);
        ls = ls * cc + e;
        o0 = o0 * cc + e * vv.x; o1 = o1 * cc + e * vv.y;
        o2 = o2 * cc + e * vv.z; o3 = o3 * cc + e * vv.w;
        m = mn;
      }
      float inv = 1.f / ls;
      float4 ov; ov.x = o0 * inv; ov.y = o1 * inv; ov.z = o2 * inv; ov.w = o3 * inv;
      *(float4*)(s_q + p * DIM + h * HD) = ov;   // overwrite own q slice
    }
    __syncthreads();

    // ---- out-proj: (288x16) @ (16x16)^T -> s_wsc ----
    for (int mt = wave; mt < 18; mt += 8) {
      int m0 = mt * 16;
      v8f acc = {0, 0, 0, 0, 0, 0, 0, 0};
#pragma unroll
      for (int k0 = 0; k0 < 16; k0 += 4)
        acc = wmma4(ld_frag(s_q, m0, k0, DIM), ld_frag(Wo, 0, k0, DIM), acc);
      int col = lane & 15;
      float bias = Bo[col];
      int rb = m0 + 8 * (lane >> 4);
#pragma unroll
      for (int g = 0; g < 8; ++g) s_wsc[(rb + g) * DIM + col] = acc[g] + bias;
    }
    __syncthreads();

    // ---- residual + LN1 + LN2 ----
    for (int p = tid; p < TC; p += NTHREADS) {
      float v[16];
#pragma unroll
      for (int c = 0; c < 16; ++c) v[c] = s_x[p * DIM + c] + s_wsc[p * DIM + c];
      ln16(v, g_l1s + l * DIM, g_l1b + l * DIM);
      ln16(v, g_l2s + l * DIM, g_l2b + l * DIM);
#pragma unroll
      for (int c = 0; c < 16; ++c) s_x[p * DIM + c] = v[c];
    }
    __syncthreads();

    // ---- FFN fused per 16-row tile: h=relu(x@w1^T+b1); y=h@w2^T+b2 ----
    {
      float* hsc = s_wsc + wave * 1024;  // private 16x64 relayout scratch
      int col = lane & 15;
      int hi = lane >> 4;
      for (int mt = wave; mt < 18; mt += 8) {
        int m0 = mt * 16;
#pragma unroll
        for (int nt = 0; nt < 4; ++nt) {
          v8f acc = {0, 0, 0, 0, 0, 0, 0, 0};
#pragma unroll
          for (int k0 = 0; k0 < 16; k0 += 4)
            acc = wmma4(ld_frag(s_x, m0, k0, DIM), ld_frag(W1, nt * 16, k0, DIM), acc);
          float bias = B1[nt * 16 + col];
#pragma unroll
          for (int g = 0; g < 8; ++g)
            hsc[(8 * hi + g) * 64 + nt * 16 + col] = fmaxf(acc[g] + bias, 0.f);
        }
        v8f acc2 = {0, 0, 0, 0, 0, 0, 0, 0};
#pragma unroll
        for (int k0 = 0; k0 < 64; k0 += 4)
          acc2 = wmma4(ld_frag(hsc, 0, k0, 64), ld_frag(W2, 0, k0, 64), acc2);
        float bias2 = B2[col];
        int rb = m0 + 8 * hi;
#pragma unroll
        for (int g = 0; g < 8; ++g) s_q[(rb + g) * DIM + col] = acc2[g] + bias2;
      }
    }
    __syncthreads();

    // ---- residual + LN3 ----
    for (int p = tid; p < TC; p += NTHREADS) {
      float v[16];
#pragma unroll
      for (int c = 0; c < 16; ++c) v[c] = s_x[p * DIM + c] + s_q[p * DIM + c];
      ln16(v, g_l3s + l * DIM, g_l3b + l * DIM);
#pragma unroll
      for (int c = 0; c < 16; ++c) s_x[p * DIM + c] = v[c];
    }
    __syncthreads();
  }

  // val[0] = proj(out[Tc-1])
  if (tid == 0) {
    float a = g_pb[0];
#pragma unroll
    for (int c = 0; c < 16; ++c) a += g_pw[c] * s_x[(TC - 1) * DIM + c];
    s_val[0] = a;
  }
  __syncthreads();

  // =================== Incremental decode, steps t = 1..11 ===================
  for (int t = 1; t < TF; ++t) {
    const int P = TC + t - 1;  // position of the token processed this step
    if (tid < 16) {
      float dv = (tid == 0) ? s_val[t - 1] : tgt[(t - 1) * DIM + tid];
      s_xin[tid] = dv + g_pos[P * DIM + tid];
    }
    __syncthreads();
    for (int l = 0; l < NL_; ++l) {
      // qkv for one token; append K,V at position P
      if (tid < 48) {
        const float* wr = g_inw + l * 48 * DIM + tid * DIM;
        float a = g_inb[l * 48 + tid];
#pragma unroll
        for (int c = 0; c < 16; ++c) a += wr[c] * s_xin[c];
        if (tid < 16)      s_q1[tid] = a;
        else if (tid < 32) s_kc[l * LPAD * DIM + P * DIM + (tid - 16)] = a;
        else               s_vc[l * LPAD * DIM + P * DIM + (tid - 32)] = a;
      }
      __syncthreads();
      // attention: wave h = head h, lanes stride keys
      if (wave < HEADS) {
        const int h = wave;
        float q0 = s_q1[h * HD + 0], q1v = s_q1[h * HD + 1];
        float q2v = s_q1[h * HD + 2], q3v = s_q1[h * HD + 3];
        const float* Kh = s_kc + l * LPAD * DIM + h * HD;
        const float* Vh = s_vc + l * LPAD * DIM + h * HD;
        float m = -1e30f, ls = 0.f, o0 = 0, o1 = 0, o2 = 0, o3 = 0;
        for (int j = lane; j <= P; j += 32) {
          float4 kv = *(const float4*)(Kh + j * DIM);
          float s = 0.5f * (q0 * kv.x + q1v * kv.y + q2v * kv.z + q3v * kv.w);
          float mn = fmaxf(m, s);
          float cc = __expf(m - mn), e = __expf(s - mn);
          float4 vv = *(const float4*)(Vh + j * DIM);
          ls = ls * cc + e;
          o0 = o0 * cc + e * vv.x; o1 = o1 * cc + e * vv.y;
          o2 = o2 * cc + e * vv.z; o3 = o3 * cc + e * vv.w;
          m = mn;
        }
#pragma unroll
        for (int off = 16; off >= 1; off >>= 1) {
          float m2 = __shfl_xor(m, off, 32);
          float l2 = __shfl_xor(ls, off, 32);
          float p0 = __shfl_xor(o0, off, 32);
          float p1 = __shfl_xor(o1, off, 32);
          float p2 = __shfl_xor(o2, off, 32);
          float p3 = __shfl_xor(o3, off, 32);
          float mn = fmaxf(m, m2);
          float cA = __expf(m - mn), cB = __expf(m2 - mn);
          ls = ls * cA + l2 * cB;
          o0 = o0 * cA + p0 * cB; o1 = o1 * cA + p1 * cB;
          o2 = o2 * cA + p2 * cB; o3 = o3 * cA + p3 * cB;
          m = mn;
        }
        if (lane == 0) {
          float inv = 1.f / ls;
          s_att[h * HD + 0] = o0 * inv; s_att[h * HD + 1] = o1 * inv;
          s_att[h * HD + 2] = o2 * inv; s_att[h * HD + 3] = o3 * inv;
        }
      }
      __syncthreads();
      // out-proj + residual
      if (tid < 16) {
        const float* wr = g_outw + l * DIM * DIM + tid * DIM;
        float a = g_outb[l * DIM + tid];
#pragma unroll
        for (int c = 0; c < 16; ++c) a += wr[c] * s_att[c];
        s_tmp[tid] = s_xin[tid] + a;
      }
      __syncthreads();
      // LN1 + LN2 (lanes 0..15 of wave 0, shuffle reductions)
      if (tid < 16) {
        float v = s_tmp[tid];
        float mean = gsum16(v) * 0.0625f;
        float d = v - mean;
        float var = gsum16(d * d) * 0.0625f;
        float x1 = d * rsqrtf(var + EPS_) * g_l1s[l * DIM + tid] + g_l1b[l * DIM + tid];
        float mean2 = gsum16(x1) * 0.0625f;
        float d2 = x1 - mean2;
        float var2 = gsum16(d2 * d2) * 0.0625f;
        s_xin[tid] = d2 * rsqrtf(var2 + EPS_) * g_l2s[l * DIM + tid] + g_l2b[l * DIM + tid];
      }
      __syncthreads();
      // FFN1
      if (tid < 64) {
        const float* wr = g_w1 + l * 64 * DIM + tid * DIM;
        float a = g_b1[l * 64 + tid];
#pragma unroll
        for (int c = 0; c < 16; ++c) a += wr[c] * s_xin[c];
        s_h64[tid] = fmaxf(a, 0.f);
      }
      __syncthreads();
      // FFN2 + residual
      if (tid < 16) {
        const float* wr = g_w2 + l * DIM * 64 + tid * 64;
        float a = g_b2[l * DIM + tid];
#pragma unroll
        for (int k = 0; k < 64; ++k) a += wr[k] * s_h64[k];
        s_tmp[tid] = s_xin[tid] + a;
      }
      __syncthreads();
      // LN3
      if (tid < 16) {
        float v = s_tmp[tid];
        float mean = gsum16(v) * 0.0625f;
        float d = v - mean;
        float var = gsum16(d * d) * 0.0625f;
        s_xin[tid] = d * rsqrtf(var + EPS_) * g_l3s[l * DIM + tid] + g_l3b[l * DIM + tid];
      }
      __syncthreads();
    }
    if (tid == 0) {
      float a = g_pb[0];
#pragma unroll
      for (int c = 0; c < 16; ++c) a += g_pw[c] * s_xin[c];
      s_val[t] = a;
    }
    __syncthreads();
  }

  // y_pred[b, t] = proj(dec[Tc+t])  where dec[Tc+t] = (val[t], tgt_future[t,1:])
  if (tid < TF) {
    float a = g_pb[0] + g_pw[0] * s_val[tid];
#pragma unroll
    for (int c = 1; c < 16; ++c) a += g_pw[c] * tgt[tid * DIM + c];
    g_out[b * TF + tid] = a;
  }
}

extern "C" void kernel_launch(void* const* d_in, const int* in_sizes, int n_in,
                              void* d_out, int out_size, void* d_ws, size_t ws_size,
                              hipStream_t stream) {
  const float* ctx  = (const float*)d_in[0];
  const float* tgtf = (const float*)d_in[1];
  const float* pos  = (const float*)d_in[2];
  const float* inw  = (const float*)d_in[3];
  const float* inb  = (const float*)d_in[4];
  const float* outw = (const float*)d_in[5];
  const float* outb = (const float*)d_in[6];
  const float* l1s  = (const float*)d_in[7];
  const float* l1b  = (const float*)d_in[8];
  const float* l2s  = (const float*)d_in[9];
  const float* l2b  = (const float*)d_in[10];
  const float* l3s  = (const float*)d_in[11];
  const float* l3b  = (const float*)d_in[12];
  const float* w1   = (const float*)d_in[13];
  const float* b1   = (const float*)d_in[14];
  const float* w2   = (const float*)d_in[15];
  const float* b2   = (const float*)d_in[16];
  const float* pw   = (const float*)d_in[17];
  const float* pb   = (const float*)d_in[18];
  float* out = (float*)d_out;

  const int Bn = in_sizes[0] / (TC * DIM);  // 256
  const size_t shmem = (size_t)TOTAL_F * sizeof(float);  // ~147 KB of the 320 KB/WGP LDS
  (void)hipFuncSetAttribute((const void*)mbar_decoder_kernel,
                            hipFuncAttributeMaxDynamicSharedMemorySize, (int)shmem);
  mbar_decoder_kernel<<<dim3(Bn), dim3(NTHREADS), shmem, stream>>>(
      ctx, tgtf, pos, inw, inb, outw, outb, l1s, l1b, l2s, l2b, l3s, l3b,
      w1, b1, w2, b2, pw, pb, out);
}